// W16A16SparseCompressedLinear_39204461478516
// MI455X (gfx1250) — compile-verified
//
#include <hip/hip_runtime.h>

// ---------------------------------------------------------------------------
// W16A16 1:2-structured-sparse linear  y = x @ W_dense^T + bias  (MI455X)
//   x[32,4096] f32, weight[8192,2048] f32 (compressed), index[8192,2048] i32,
//   bias[8192] f32 -> y[32,8192] f32
//
// Roofline: weight+index stream = 128MB from HBM -> ~5.5us @ 23.3TB/s floor.
// V_SWMMAC_F32_16X16X64_BF16 consumes the 1:2 sparsity natively (group codes
// idx0=bit(pair0) in {0,1}, idx1=2+bit(pair1) in {2,3}; idx0<idx1 holds).
// Phase 1 pre-packs x to bf16 in d_ws (256KB, L2-resident) so the hot loop's
// B fragments are straight b128 loads with zero conversion VALU.
// 8-way K split per 16-row tile -> 4096 waves for stream latency hiding;
// deterministic LDS tree reduction + bias at the end.
// ---------------------------------------------------------------------------

typedef __attribute__((ext_vector_type(8)))  float  v8f;
typedef __attribute__((ext_vector_type(16))) __bf16 v16bf;
typedef __attribute__((ext_vector_type(32))) __bf16 v32bf;
typedef __attribute__((ext_vector_type(2)))  __bf16 v2bf;

#define TOKENS 32
#define IN_F   4096
#define OUT_F  8192
#define NNZ    (IN_F / 2)
#define KWAVES 8                 // K-split factor (waves per block)
#define KCHUNK (IN_F / KWAVES)   // 512 dense k per wave -> 8 swmmac steps

// ---- f32 pair -> packed bf16x2 (RNE), pure 32-bit integer form ----
__device__ __forceinline__ unsigned int pk2bf(float fa, float fb) {
#if __has_builtin(__builtin_amdgcn_cvt_pk_bf16_f32)
  v2bf r = __builtin_amdgcn_cvt_pk_bf16_f32(fa, fb);
  return __builtin_bit_cast(unsigned int, r);
#else
  unsigned int a = __builtin_bit_cast(unsigned int, fa);
  unsigned int b = __builtin_bit_cast(unsigned int, fb);
  a += 0x7FFFu + ((a >> 16) & 1u);          // v_bfe + v_add3
  b += 0x7FFFu + ((b >> 16) & 1u);
  return (a >> 16) | (b & 0xFFFF0000u);     // v_lshr + v_and_or
#endif
}

__device__ __forceinline__ void load8f(float* d, const float* s) {
  *(float4*)(d + 0) = *(const float4*)(s + 0);
  *(float4*)(d + 4) = *(const float4*)(s + 4);
}
__device__ __forceinline__ void load16f(float* d, const float* s) {
  *(float4*)(d + 0)  = *(const float4*)(s + 0);
  *(float4*)(d + 4)  = *(const float4*)(s + 4);
  *(float4*)(d + 8)  = *(const float4*)(s + 8);
  *(float4*)(d + 12) = *(const float4*)(s + 12);
}
__device__ __forceinline__ void load8i(int* d, const int* s) {
  *(int4*)(d + 0) = *(const int4*)(s + 0);
  *(int4*)(d + 4) = *(const int4*)(s + 4);
}

// ---- phase 1: x f32 -> bf16 (row-major [32][4096] bf16, as packed u32) ----
__global__ __launch_bounds__(256) void pack_x_bf16(
    const float* __restrict__ x, unsigned int* __restrict__ xbf) {
  const int i = blockIdx.x * 256 + threadIdx.x;  // one packed u32 each
  if (i < (TOKENS * IN_F) / 2) {
    float2 f = *(const float2*)(x + 2 * i);
    xbf[i] = pk2bf(f.x, f.y);
  }
}

// ---- phase 2: sparse GEMM via SWMMAC ----
template <bool PREPACKED>
__global__ __launch_bounds__(32 * KWAVES) void sparse_linear_swmmac(
    const float* __restrict__ x, const unsigned int* __restrict__ xbf,
    const float* __restrict__ w, const int* __restrict__ sidx,
    const float* __restrict__ bias, float* __restrict__ y) {
  __shared__ float red[KWAVES][2][8][32];  // [wave][tok-half][c-vgpr][lane]

  const int lane = threadIdx.x & 31;
  const int wave = __builtin_amdgcn_readfirstlane((int)(threadIdx.x >> 5));
  const int hi16 = (lane >> 4) & 1;  // lane half per ISA fragment layouts
  const int m    = lane & 15;        // row-in-tile / token-in-half
  const int o    = blockIdx.x * 16 + m;

  const float*        xr0  = x   + (size_t)m * IN_F;
  const float*        xr1  = x   + (size_t)(m + 16) * IN_F;
  const unsigned int* xb0  = xbf + (size_t)m * (IN_F / 2);
  const unsigned int* xb1  = xbf + (size_t)(m + 16) * (IN_F / 2);
  const float*        wrow = w    + (size_t)o * NNZ;
  const int*          irow = sidx + (size_t)o * NNZ;

  v8f c0 = {};  // D tile, tokens 0-15
  v8f c1 = {};  // D tile, tokens 16-31

  const int kBeg = wave * KCHUNK;  // scalar loop bounds
  const int kEnd = kBeg + KCHUNK;

  for (int kb = kBeg; kb < kEnd; kb += 64) {
    const int cb = kb >> 1;  // compressed-column base of this 64-k chunk

    // software prefetch the HBM streams ~2 iterations ahead
    __builtin_prefetch(wrow + cb + 64, 0, 1);
    __builtin_prefetch(irow + cb + 64, 0, 1);

    // ---- A fragment (stored 16x32 bf16): lanes 0-15 hold stored cols
    //      {0-7,16-23}, lanes 16-31 hold {8-15,24-31}.
    float av[16];
    load8f(av + 0, wrow + cb + (hi16 ? 8 : 0));
    load8f(av + 8, wrow + cb + (hi16 ? 8 : 0) + 16);
    unsigned int au[8];
#pragma unroll
    for (int e = 0; e < 8; ++e) au[e] = pk2bf(av[2 * e], av[2 * e + 1]);
    v16bf aV;
    __builtin_memcpy(&aV, au, 32);

    // ---- sparse index fragment: lanes 0-15 -> dense k 0-31 of chunk,
    //      lanes 16-31 -> dense k 32-63; group g: idx0=bit(2g), idx1=2|bit(2g+1)
    int iv[16];
    load8i(iv + 0, irow + cb + (hi16 ? 16 : 0));
    load8i(iv + 8, irow + cb + (hi16 ? 16 : 0) + 8);
    unsigned int pk = 0;
#pragma unroll
    for (int g = 0; g < 8; ++g) {
      unsigned int code =
          (unsigned)(iv[2 * g] & 1) | ((2u | (unsigned)(iv[2 * g + 1] & 1)) << 2);
      pk |= code << (4 * g);
    }

    // ---- B fragments (dense 64x16 bf16 = x^T): per lane, elements 0..15 =
    //      k kk..kk+15, elements 16..31 = k kk+32..kk+47, kk = kb + 16*hi16.
    const int kk = kb + (hi16 ? 16 : 0);
    v32bf b0V, b1V;
    if constexpr (PREPACKED) {
      const int cu = kk >> 1;  // packed-u32 column
      unsigned int bu0[16], bu1[16];
      *(uint4*)(bu0 + 0)  = *(const uint4*)(xb0 + cu);
      *(uint4*)(bu0 + 4)  = *(const uint4*)(xb0 + cu + 4);
      *(uint4*)(bu0 + 8)  = *(const uint4*)(xb0 + cu + 16);
      *(uint4*)(bu0 + 12) = *(const uint4*)(xb0 + cu + 20);
      *(uint4*)(bu1 + 0)  = *(const uint4*)(xb1 + cu);
      *(uint4*)(bu1 + 4)  = *(const uint4*)(xb1 + cu + 4);
      *(uint4*)(bu1 + 8)  = *(const uint4*)(xb1 + cu + 16);
      *(uint4*)(bu1 + 12) = *(const uint4*)(xb1 + cu + 20);
      __builtin_memcpy(&b0V, bu0, 64);
      __builtin_memcpy(&b1V, bu1, 64);
    } else {
      float b0f[32], b1f[32];
      load16f(b0f + 0, xr0 + kk);
      load16f(b0f + 16, xr0 + kk + 32);
      load16f(b1f + 0, xr1 + kk);
      load16f(b1f + 16, xr1 + kk + 32);
      unsigned int bu0[16], bu1[16];
#pragma unroll
      for (int e = 0; e < 16; ++e) {
        bu0[e] = pk2bf(b0f[2 * e], b0f[2 * e + 1]);
        bu1[e] = pk2bf(b1f[2 * e], b1f[2 * e + 1]);
      }
      __builtin_memcpy(&b0V, bu0, 64);
      __builtin_memcpy(&b1V, bu1, 64);
    }

    c0 = __builtin_amdgcn_swmmac_f32_16x16x64_bf16(false, aV, false, b0V, c0,
                                                   (int)pk, false, false);
    c1 = __builtin_amdgcn_swmmac_f32_16x16x64_bf16(false, aV, false, b1V, c1,
                                                   (int)pk, false, false);
  }

  // ---- deterministic cross-wave K reduction + bias + store ----
#pragma unroll
  for (int r = 0; r < 8; ++r) {
    red[wave][0][r][lane] = c0[r];
    red[wave][1][r][lane] = c1[r];
  }
  __syncthreads();
  for (int i = threadIdx.x; i < 512; i += 32 * KWAVES) {
    const int l = i & 31;
    const int r = (i >> 5) & 7;
    const int h = (i >> 8) & 1;
    float s = 0.0f;
#pragma unroll
    for (int wv = 0; wv < KWAVES; ++wv) s += red[wv][h][r][l];
    // 32-bit C/D 16x16 layout: lanes 0-15 -> M=r, lanes 16-31 -> M=8+r; N=l%16
    const int mm = (l < 16) ? r : (r + 8);
    const int n  = (l & 15) + 16 * h;
    const int oo = blockIdx.x * 16 + mm;
    y[(size_t)n * OUT_F + oo] = s + bias[oo];
  }
}

extern "C" void kernel_launch(void* const* d_in, const int* in_sizes, int n_in,
                              void* d_out, int out_size, void* d_ws,
                              size_t ws_size, hipStream_t stream) {
  (void)in_sizes; (void)n_in; (void)out_size;
  const float* x    = (const float*)d_in[0];
  const float* w    = (const float*)d_in[1];
  const int*   sidx = (const int*)d_in[2];
  const float* bias = (const float*)d_in[3];
  float*       y    = (float*)d_out;

  const size_t need = (size_t)TOKENS * IN_F * 2;  // 256KB bf16 copy of x
  if (d_ws != nullptr && ws_size >= need) {
    unsigned int* xbf = (unsigned int*)d_ws;
    pack_x_bf16<<<dim3((TOKENS * IN_F / 2) / 256), dim3(256), 0, stream>>>(x, xbf);
    sparse_linear_swmmac<true><<<dim3(OUT_F / 16), dim3(32 * KWAVES), 0, stream>>>(
        x, xbf, w, sidx, bias, y);
  } else {
    sparse_linear_swmmac<false><<<dim3(OUT_F / 16), dim3(32 * KWAVES), 0, stream>>>(
        x, nullptr, w, sidx, bias, y);
  }
}